// Network_61314953117854
// MI455X (gfx1250) — compile-verified
//
#include <hip/hip_runtime.h>
#include <cstdint>
#include <cstddef>

// Problem constants (match reference)
constexpr int   S_SYN   = 32768;
constexpr int   G_GEN   = 512;
constexpr int   T_STEPS = 1000;
constexpr float DT_MS   = 0.1f;

// Main-kernel tiling
constexpr int BLK          = 128;                          // threads per block (4 wave32)
constexpr int CHUNK        = 8;                            // timesteps staged per async phase
constexpr int STAGE_ISSUES = (CHUNK * G_GEN) / (BLK * 4);  // b128 issues per thread = 8
constexpr int NPHASES      = (T_STEPS - 1 + CHUNK - 1) / CHUNK;  // 125

// ---------------------------------------------------------------------------
// Modified Bessel I0, Abramowitz & Stegun single-precision polynomial approx.
// Overflows to +inf for large x the same way jax.scipy.special.i0 (f32) does.
// ---------------------------------------------------------------------------
__device__ __forceinline__ float bessel_i0f(float x) {
    float ax = fabsf(x);
    if (ax < 3.75f) {
        float y = x / 3.75f;
        y *= y;
        return 1.0f + y * (3.5156229f + y * (3.0899424f + y * (1.2067492f
                 + y * (0.2659732f + y * (0.0360768f + y * 0.0045813f)))));
    } else {
        float y = 3.75f / ax;
        return (expf(ax) * rsqrtf(ax)) *
               (0.39894228f + y * (0.01328592f + y * (0.00225319f
                 + y * (-0.00157565f + y * (0.00916281f + y * (-0.02057706f
                 + y * (0.02635537f + y * (-0.01647633f + y * 0.00392377f))))))));
    }
}

__device__ __forceinline__ float kappa_from_R(float R) {
    // piecewise r2kappa from reference
    if (R < 0.53f) {
        float R2 = R * R;
        return 2.0f * R + R * R2 + (5.0f / 6.0f) * R * R2 * R2;
    } else if (R < 0.85f) {
        return -0.4f + 1.39f * R + 0.43f / (1.0f - R);
    } else {
        return 1.0f / (3.0f * R - 4.0f * R * R + R * R * R);
    }
}

// ---------------------------------------------------------------------------
// Kernel A: fir[T][G] table + per-step Euler scale rscale[s] = (t[s]-t[s-1])/DT.
// 512K threads; transcendental cost shared across all synapses (512K exp/cos
// instead of 32.7M if computed per synapse). Table = 2 MB, stays hot in L2.
// ---------------------------------------------------------------------------
__global__ void __launch_bounds__(256)
fir_table_kernel(const float* __restrict__ R,
                 const float* __restrict__ omega,
                 const float* __restrict__ phase,
                 const float* __restrict__ msr,
                 const float* __restrict__ t,
                 float* __restrict__ fir_tab,   // [T_STEPS * G_GEN]
                 float* __restrict__ rscale)    // [T_STEPS]
{
    int id = blockIdx.x * blockDim.x + threadIdx.x;
    if (id >= T_STEPS * G_GEN) return;
    int s = id >> 9;            // id / G_GEN
    int g = id & (G_GEN - 1);   // id % G_GEN

    float kap  = kappa_from_R(R[g]);
    float norm = msr[g] / bessel_i0f(kap) * 0.001f;
    float m4t  = 2.0f * 3.14159265358979323846f * omega[g] * 0.001f;
    float ti   = t[s];
    fir_tab[id] = norm * expf(kap * cosf(m4t * ti - phase[g]));

    if (g == 0) {
        rscale[s] = (s > 0) ? (t[s] - t[s - 1]) * (1.0f / DT_MS) : 0.0f;
    }
}

// ---------------------------------------------------------------------------
// CDNA5 async-to-LDS helpers (ASYNCcnt path; inline asm is the portable route).
// Flat shared-variable pointers carry the LDS byte offset in their low 32 bits.
// ---------------------------------------------------------------------------
__device__ __forceinline__ void async_stage_b128(const float* gsrc, const void* lds_dst) {
    unsigned lds_off = (unsigned)(uintptr_t)lds_dst;
    asm volatile("global_load_async_to_lds_b128 %0, %1, off"
                 :: "v"(lds_off), "v"(gsrc)
                 : "memory");
}

// ---------------------------------------------------------------------------
// Kernel B: one lane per synapse, sequential over T. fir rows are staged in
// 8-row chunks (16 KB) into a triple-buffered LDS ring via
// global_load_async_to_lds_b128:
//   - one s_wait_asynccnt + one workgroup barrier per 8 timesteps (125 phases)
//   - inner 8-step loop is sync-free: ds_load gather + FLOPs + NT stores,
//     freely pipelined by the compiler
//   - triple buffering removes the trailing barrier (buffer for phase p+1 was
//     last read in phase p-2; the top-of-phase barrier separates them)
// Output (393 MB > 192 MB L2) is streamed with non-temporal stores.
// ---------------------------------------------------------------------------
__global__ void __launch_bounds__(BLK)
synapse_table_kernel(const float* __restrict__ tau_d,
                     const float* __restrict__ tau_r,
                     const float* __restrict__ tau_f,
                     const float* __restrict__ Uinc,
                     const float* __restrict__ W,
                     const int*   __restrict__ pre_idx,
                     const float* __restrict__ fir_tab,
                     const float* __restrict__ rscale,
                     float* __restrict__ out)       // [T_STEPS][3*S_SYN]
{
    __shared__ __align__(16) float lds_fir[3][CHUNK * G_GEN];   // 3 x 16 KB

    const int tid = threadIdx.x;
    const int sid = blockIdx.x * BLK + tid;

    // Per-synapse constants (loaded once, register-resident for the whole run)
    const float td = tau_d[sid];
    const float tr = tau_r[sid];
    const float tf = tau_f[sid];
    const float ui = Uinc[sid];
    const float w  = W[sid];
    const int   pre = pre_idx[sid];
    const float tau1r = (td != tr) ? td / (td - tr) : 1e-13f;
    const float ed = expf(-DT_MS / td);
    const float er = expf(-DT_MS / tr);
    const float ef = expf(-DT_MS / tf);

    float X = 1.0f, Y = 0.0f, U = 0.0f;

    // Row 0 = initial state
    __builtin_nontemporal_store(1.0f, out + sid);
    __builtin_nontemporal_store(0.0f, out + S_SYN + sid);
    __builtin_nontemporal_store(0.0f, out + 2 * S_SYN + sid);

    // Stage rows [row0, row0+CHUNK) of fir_tab into lds_fir[buf].
    // 8 issues x 128 lanes x 16 B = 16 KB. Clamp tail-phase overread.
    auto stage = [&](int row0, int buf) {
#pragma unroll
        for (int i = 0; i < STAGE_ISSUES; ++i) {
            const int off = i * (BLK * 4) + 4 * tid;
            size_t gidx = (size_t)row0 * G_GEN + (size_t)off;
            if (gidx > (size_t)(T_STEPS * G_GEN - 4)) gidx = (size_t)(T_STEPS * G_GEN - 4);
            async_stage_b128(fir_tab + gidx, &lds_fir[buf][off]);
        }
    };

    // Prologue: stage phase 0 (rows 1..8) into buffer 0.
    stage(1, 0);

    for (int p = 0; p < NPHASES; ++p) {
        const int s0 = 1 + p * CHUNK;
        if (p + 1 < NPHASES) {
            // Prefetch next phase, then wait for this phase's 8 issues
            // (in-order completion: <=8 outstanding means phase p landed).
            stage(s0 + CHUNK, (p + 1) % 3);
            asm volatile("s_wait_asynccnt 0x8" ::: "memory");
        } else {
            asm volatile("s_wait_asynccnt 0x0" ::: "memory");
        }
        __syncthreads();   // all 4 waves' portions of this chunk are in LDS

        const float* buf = lds_fir[p % 3];
#pragma unroll
        for (int j = 0; j < CHUNK; ++j) {
            const int s = s0 + j;
            if (s >= T_STEPS) break;

            const float fir = buf[j * G_GEN + pre];
            const float r   = rscale[s];            // dti/DT (uniform)

            // Tsodyks-Markram update (matches reference expression order)
            const float Spre = fir * w;
            const float y_   = Y * ed;
            const float tY   = tau1r * Y;
            const float x_   = 1.0f + (X - 1.0f + tY) * er - tY;
            const float u_   = U * ef;
            const float u0   = u_ + ui * (1.0f - u_) * Spre;
            const float a    = u0 * x_ * Spre;
            const float y0v  = y_ + a;
            const float x0   = x_ - a;
            X = fmaf(r, x0  - X, X);
            Y = fmaf(r, y0v - Y, Y);
            U = fmaf(r, u0  - U, U);

            float* row = out + (size_t)s * (3 * S_SYN);
            __builtin_nontemporal_store(X, row + sid);
            __builtin_nontemporal_store(Y, row + S_SYN + sid);
            __builtin_nontemporal_store(U, row + 2 * S_SYN + sid);
        }
        // No trailing barrier needed: triple buffering gives >=1 barrier
        // between the last read of a buffer and its re-targeting.
    }
}

// ---------------------------------------------------------------------------
// Fallback (workspace too small): fully fused, each lane recomputes fir for
// its own generator (999 exp+cos per lane). Self-contained.
// ---------------------------------------------------------------------------
__global__ void __launch_bounds__(256)
synapse_fused_kernel(const float* __restrict__ tau_d,
                     const float* __restrict__ tau_r,
                     const float* __restrict__ tau_f,
                     const float* __restrict__ Uinc,
                     const float* __restrict__ W,
                     const float* __restrict__ R,
                     const float* __restrict__ omega,
                     const float* __restrict__ phase,
                     const float* __restrict__ msr,
                     const float* __restrict__ t,
                     const int*   __restrict__ pre_idx,
                     float* __restrict__ out)
{
    const int sid = blockIdx.x * 256 + threadIdx.x;

    const float td = tau_d[sid];
    const float tr = tau_r[sid];
    const float tf = tau_f[sid];
    const float ui = Uinc[sid];
    const float w  = W[sid];
    const int   pre = pre_idx[sid];
    const float tau1r = (td != tr) ? td / (td - tr) : 1e-13f;
    const float ed = expf(-DT_MS / td);
    const float er = expf(-DT_MS / tr);
    const float ef = expf(-DT_MS / tf);

    const float kap  = kappa_from_R(R[pre]);
    const float norm = msr[pre] / bessel_i0f(kap) * 0.001f;
    const float m4t  = 2.0f * 3.14159265358979323846f * omega[pre] * 0.001f;
    const float ph   = phase[pre];

    float X = 1.0f, Y = 0.0f, U = 0.0f;

    __builtin_nontemporal_store(1.0f, out + sid);
    __builtin_nontemporal_store(0.0f, out + S_SYN + sid);
    __builtin_nontemporal_store(0.0f, out + 2 * S_SYN + sid);

    float tprev = t[0];
    for (int s = 1; s < T_STEPS; ++s) {
        const float ti = t[s];
        const float r  = (ti - tprev) * (1.0f / DT_MS);
        tprev = ti;

        const float fir  = norm * expf(kap * cosf(m4t * ti - ph));
        const float Spre = fir * w;
        const float y_   = Y * ed;
        const float tY   = tau1r * Y;
        const float x_   = 1.0f + (X - 1.0f + tY) * er - tY;
        const float u_   = U * ef;
        const float u0   = u_ + ui * (1.0f - u_) * Spre;
        const float a    = u0 * x_ * Spre;
        const float y0v  = y_ + a;
        const float x0   = x_ - a;
        X = fmaf(r, x0  - X, X);
        Y = fmaf(r, y0v - Y, Y);
        U = fmaf(r, u0  - U, U);

        float* row = out + (size_t)s * (3 * S_SYN);
        __builtin_nontemporal_store(X, row + sid);
        __builtin_nontemporal_store(Y, row + S_SYN + sid);
        __builtin_nontemporal_store(U, row + 2 * S_SYN + sid);
    }
}

// ---------------------------------------------------------------------------
// Host launcher
// Input order (setup_inputs): tau_d, tau_r, tau_f, Uinc, W, R, omega, phase,
//                             mean_spike_rate, t, pre_idx
// ---------------------------------------------------------------------------
extern "C" void kernel_launch(void* const* d_in, const int* in_sizes, int n_in,
                              void* d_out, int out_size, void* d_ws, size_t ws_size,
                              hipStream_t stream) {
    const float* tau_d   = (const float*)d_in[0];
    const float* tau_r   = (const float*)d_in[1];
    const float* tau_f   = (const float*)d_in[2];
    const float* Uinc    = (const float*)d_in[3];
    const float* W       = (const float*)d_in[4];
    const float* R       = (const float*)d_in[5];
    const float* omega   = (const float*)d_in[6];
    const float* phase   = (const float*)d_in[7];
    const float* msr     = (const float*)d_in[8];
    const float* t       = (const float*)d_in[9];
    const int*   pre_idx = (const int*)d_in[10];
    float*       out     = (float*)d_out;

    (void)in_sizes; (void)n_in; (void)out_size;

    const size_t fir_bytes = (size_t)T_STEPS * G_GEN * sizeof(float);
    const size_t rs_bytes  = (size_t)T_STEPS * sizeof(float);

    if (ws_size >= fir_bytes + rs_bytes) {
        float* fir_tab = (float*)d_ws;
        float* rscale  = (float*)((char*)d_ws + fir_bytes);

        fir_table_kernel<<<(T_STEPS * G_GEN) / 256, 256, 0, stream>>>(
            R, omega, phase, msr, t, fir_tab, rscale);

        synapse_table_kernel<<<S_SYN / BLK, BLK, 0, stream>>>(
            tau_d, tau_r, tau_f, Uinc, W, pre_idx, fir_tab, rscale, out);
    } else {
        synapse_fused_kernel<<<S_SYN / 256, 256, 0, stream>>>(
            tau_d, tau_r, tau_f, Uinc, W, R, omega, phase, msr, t, pre_idx, out);
    }
}